// Net_14147622273471
// MI455X (gfx1250) — compile-verified
//
#include <hip/hip_runtime.h>
#include <hip/hip_bf16.h>
#include <math.h>

// -------- problem constants (match reference) --------
#define NFEAT 256
#define HID0  64
#define HID1  32
#define HID2  16
#define NCLS  4

typedef float v2f __attribute__((ext_vector_type(2)));
typedef float v8f __attribute__((ext_vector_type(8)));

// ---------------------------------------------------------------------------
// 0) init workspace: deg = 1.0 (self loop), agg = 0
// ---------------------------------------------------------------------------
__global__ void init_ws_kernel(float* __restrict__ deg, float* __restrict__ agg, int n) {
  long long i = (long long)blockIdx.x * blockDim.x + threadIdx.x;
  if (i < n) deg[i] = 1.0f;
  long long tot = (long long)n * HID0;
  if (i < tot) agg[i] = 0.0f;
}

// ---------------------------------------------------------------------------
// 1) degree: deg[dst] += 1 per edge (f32 atomics, resolved in L2)
// ---------------------------------------------------------------------------
__global__ void degree_kernel(const long long* __restrict__ ei, float* __restrict__ deg, int E) {
  int e = blockIdx.x * blockDim.x + threadIdx.x;
  if (e < E) {
    int dst = (int)ei[E + e];
    atomicAdd(&deg[dst], 1.0f);
  }
}

// ---------------------------------------------------------------------------
// 2) dinv = rsqrt(deg) in place (deg >= 1 always, so no zero branch needed)
// ---------------------------------------------------------------------------
__global__ void dinv_kernel(float* __restrict__ deg, int n) {
  int i = blockIdx.x * blockDim.x + threadIdx.x;
  if (i < n) deg[i] = rsqrtf(deg[i]);
}

// ---------------------------------------------------------------------------
// 3) H = X @ W_gcn  [n x 256] @ [256 x 64] via V_WMMA_F32_16X16X4_F32
//    - one wave computes a 16x64 row block (4 accumulator tiles of 16x16)
//    - W (256x64 f32 = 64 KB) staged in LDS, A fragments streamed from global
//    ISA f32 fragment layout (7.12.2): lanes 0-15 / 16-31 split K in halves:
//      A: lane L -> row = L&15; VGPR0 = X[row][k + 2*(L>>4)], VGPR1 = +1
//      B: lane L -> col = L&15; VGPR0 = W[k + 2*(L>>4)][col], VGPR1 = +1
//      C: VGPR v, lane L -> row v + 8*(L>>4), col L&15
// ---------------------------------------------------------------------------
__global__ void gemm_xw_kernel(const float* __restrict__ X, const float* __restrict__ W,
                               float* __restrict__ H, int nrows) {
  __shared__ float Wl[NFEAT * HID0];   // 64 KB
  for (int i = threadIdx.x; i < NFEAT * HID0; i += blockDim.x) Wl[i] = W[i];
  __syncthreads();

  const int wave = threadIdx.x >> 5;
  const int lane = threadIdx.x & 31;
  const int half = lane >> 4;     // 0 or 1  (K sub-offset 0 or 2)
  const int l15  = lane & 15;     // row (A) / col (B,C,D)

  const int row0 = (blockIdx.x * (blockDim.x >> 5) + wave) * 16;
  if (row0 >= nrows) return;

  // clamp for a (nonexistent with n=100000, but safe) ragged tail; stores are guarded
  const int arow = (row0 + l15 < nrows) ? (row0 + l15) : (nrows - 1);
  const float* __restrict__ xrow = X + (size_t)arow * NFEAT;

  v8f acc0 = {}, acc1 = {}, acc2 = {}, acc3 = {};
  for (int k = 0; k < NFEAT; k += 4) {
    v2f a;
    a.x = xrow[k + 2 * half + 0];
    a.y = xrow[k + 2 * half + 1];
    const float* w0 = &Wl[(k + 2 * half + 0) * HID0 + l15];
    const float* w1 = &Wl[(k + 2 * half + 1) * HID0 + l15];
    v2f b;
    b.x = w0[0];  b.y = w1[0];
    acc0 = __builtin_amdgcn_wmma_f32_16x16x4_f32(false, a, false, b, (short)0, acc0, false, false);
    b.x = w0[16]; b.y = w1[16];
    acc1 = __builtin_amdgcn_wmma_f32_16x16x4_f32(false, a, false, b, (short)0, acc1, false, false);
    b.x = w0[32]; b.y = w1[32];
    acc2 = __builtin_amdgcn_wmma_f32_16x16x4_f32(false, a, false, b, (short)0, acc2, false, false);
    b.x = w0[48]; b.y = w1[48];
    acc3 = __builtin_amdgcn_wmma_f32_16x16x4_f32(false, a, false, b, (short)0, acc3, false, false);
  }

#pragma unroll
  for (int v = 0; v < 8; ++v) {
    int r = row0 + v + half * 8;
    if (r < nrows) {
      float* hr = H + (size_t)r * HID0 + l15;
      hr[0]  = acc0[v];
      hr[16] = acc1[v];
      hr[32] = acc2[v];
      hr[48] = acc3[v];
    }
  }
}

// ---------------------------------------------------------------------------
// 4) edge scatter: agg[dst] += h[src] * dinv[src]*dinv[dst]
//    one wave per edge; lanes share (src,dst) -> scalar broadcast loads;
//    each lane moves a float2 slice of the 64-feature message.
// ---------------------------------------------------------------------------
__global__ void scatter_kernel(const long long* __restrict__ ei,
                               const float* __restrict__ h,
                               const float* __restrict__ dinv,
                               float* __restrict__ agg, int E) {
  long long tid = (long long)blockIdx.x * blockDim.x + threadIdx.x;
  long long e = tid >> 5;
  int lane = (int)(tid & 31);
  if (e >= E) return;
  int src = (int)ei[e];
  int dst = (int)ei[E + e];
  float norm = dinv[src] * dinv[dst];
  const float2* __restrict__ hs = (const float2*)(h + (size_t)src * HID0);
  float2 v = hs[lane];
  float* ad = agg + (size_t)dst * HID0 + 2 * lane;
  atomicAdd(ad + 0, v.x * norm);
  atomicAdd(ad + 1, v.y * norm);
}

// ---------------------------------------------------------------------------
// 5) post: fold self-loop analytically, add bias, relu (in place on agg)
//    agg[i][f] = relu(agg[i][f] + h[i][f]*dinv[i]^2 + b[f])
// ---------------------------------------------------------------------------
__global__ void post_kernel(const float* __restrict__ h, const float* __restrict__ dinv,
                            const float* __restrict__ b, float* __restrict__ agg, int n) {
  long long i = (long long)blockIdx.x * blockDim.x + threadIdx.x;
  if (i >= (long long)n * HID0) return;
  int node = (int)(i >> 6);
  int f = (int)(i & (HID0 - 1));
  float d = dinv[node];
  float v = agg[i] + h[i] * d * d + b[f];
  agg[i] = v > 0.0f ? v : 0.0f;
}

// ---------------------------------------------------------------------------
// 6) MLP + log_softmax: one thread per node, weights in LDS (~10.7 KB)
// ---------------------------------------------------------------------------
__global__ void mlp_kernel(const float* __restrict__ hpost,
                           const float* __restrict__ W1, const float* __restrict__ b1,
                           const float* __restrict__ W2, const float* __restrict__ b2,
                           const float* __restrict__ W3, const float* __restrict__ b3,
                           float* __restrict__ out, int n) {
  __shared__ float sW1[HID0 * HID1];
  __shared__ float sW2[HID1 * HID2];
  __shared__ float sW3[HID2 * NCLS];
  __shared__ float sb1[HID1], sb2[HID2], sb3[NCLS];
  for (int i = threadIdx.x; i < HID0 * HID1; i += blockDim.x) sW1[i] = W1[i];
  for (int i = threadIdx.x; i < HID1 * HID2; i += blockDim.x) sW2[i] = W2[i];
  for (int i = threadIdx.x; i < HID2 * NCLS; i += blockDim.x) sW3[i] = W3[i];
  if (threadIdx.x < HID1) sb1[threadIdx.x] = b1[threadIdx.x];
  if (threadIdx.x < HID2) sb2[threadIdx.x] = b2[threadIdx.x];
  if (threadIdx.x < NCLS) sb3[threadIdx.x] = b3[threadIdx.x];
  __syncthreads();

  int node = blockIdx.x * blockDim.x + threadIdx.x;
  if (node >= n) return;

  float hv[HID0];
  const float4* hp = (const float4*)(hpost + (size_t)node * HID0);
#pragma unroll
  for (int v = 0; v < HID0 / 4; ++v) {
    float4 t = hp[v];
    hv[4 * v + 0] = t.x; hv[4 * v + 1] = t.y; hv[4 * v + 2] = t.z; hv[4 * v + 3] = t.w;
  }

  float h1[HID1];
#pragma unroll 4
  for (int j = 0; j < HID1; ++j) {
    float acc = sb1[j];
    for (int k = 0; k < HID0; ++k) acc = fmaf(hv[k], sW1[k * HID1 + j], acc);
    h1[j] = acc > 0.0f ? acc : 0.0f;
  }

  float h2[HID2];
#pragma unroll 4
  for (int j = 0; j < HID2; ++j) {
    float acc = sb2[j];
    for (int k = 0; k < HID1; ++k) acc = fmaf(h1[k], sW2[k * HID2 + j], acc);
    h2[j] = acc > 0.0f ? acc : 0.0f;
  }

  float z[NCLS];
#pragma unroll
  for (int j = 0; j < NCLS; ++j) {
    float acc = sb3[j];
    for (int k = 0; k < HID2; ++k) acc = fmaf(h2[k], sW3[k * NCLS + j], acc);
    z[j] = acc;
  }

  float m = z[0];
#pragma unroll
  for (int j = 1; j < NCLS; ++j) m = z[j] > m ? z[j] : m;
  float s = 0.0f;
#pragma unroll
  for (int j = 0; j < NCLS; ++j) s += __expf(z[j] - m);
  float lse = m + __logf(s);
  float* o = out + (size_t)node * NCLS;
#pragma unroll
  for (int j = 0; j < NCLS; ++j) o[j] = z[j] - lse;
}

// ---------------------------------------------------------------------------
// launcher
// ---------------------------------------------------------------------------
extern "C" void kernel_launch(void* const* d_in, const int* in_sizes, int n_in,
                              void* d_out, int out_size, void* d_ws, size_t ws_size,
                              hipStream_t stream) {
  const float*     x   = (const float*)d_in[0];
  const long long* ei  = (const long long*)d_in[1];
  const float*     Wg  = (const float*)d_in[2];
  const float*     bg  = (const float*)d_in[3];
  const float*     W1  = (const float*)d_in[4];
  const float*     b1  = (const float*)d_in[5];
  const float*     W2  = (const float*)d_in[6];
  const float*     b2  = (const float*)d_in[7];
  const float*     W3  = (const float*)d_in[8];
  const float*     b3  = (const float*)d_in[9];

  const int n = in_sizes[0] / NFEAT;     // 100000
  const int E = in_sizes[1] / 2;         // 3200000

  // workspace layout (floats): [deg/dinv : n][h : n*64][agg : n*64] (~51.6 MB)
  float* deg = (float*)d_ws;
  float* h   = deg + n;
  float* agg = h + (size_t)n * HID0;

  const int TB = 256;
  long long featTot = (long long)n * HID0;

  // 0) init deg=1, agg=0
  {
    unsigned int g = (unsigned int)((featTot + TB - 1) / TB);
    init_ws_kernel<<<g, TB, 0, stream>>>(deg, agg, n);
  }
  // 1) degrees
  degree_kernel<<<(E + TB - 1) / TB, TB, 0, stream>>>(ei, deg, E);
  // 2) dinv
  dinv_kernel<<<(n + TB - 1) / TB, TB, 0, stream>>>(deg, n);
  // 3) WMMA GEMM: 8 waves/block -> 128 rows/block
  {
    unsigned int g = (unsigned int)((n + 127) / 128);
    gemm_xw_kernel<<<g, TB, 0, stream>>>(x, Wg, h, n);
  }
  // 4) edge scatter: one wave per edge
  {
    long long threads = (long long)E * 32;
    unsigned int g = (unsigned int)((threads + TB - 1) / TB);
    scatter_kernel<<<g, TB, 0, stream>>>(ei, h, deg, agg, E);
  }
  // 5) self-loop + bias + relu
  {
    unsigned int g = (unsigned int)((featTot + TB - 1) / TB);
    post_kernel<<<g, TB, 0, stream>>>(h, deg, bg, agg, n);
  }
  // 6) MLP + log_softmax
  mlp_kernel<<<(n + TB - 1) / TB, TB, 0, stream>>>(agg, W1, b1, W2, b2, W3, b3,
                                                   (float*)d_out, n);
}